// SpatialAtt_24094766531085
// MI455X (gfx1250) — compile-verified
//
#include <hip/hip_runtime.h>
#include <hip/hip_bf16.h>

// SpatialAtt for MI455X (gfx1250).
//
// Rank-1 softmax attention collapsed to per-row exp-weighted reductions;
// the P@[att, 1] reduction runs on the matrix pipe via V_WMMA_F32_16X16X4_F32
// (flash-attention style: one WMMA accumulates numerator and denominator for
// a 16-row tile and 4 columns at once).
//
// HBM traffic ~100 MB (x read twice + out written once) -> ~4.3us floor at
// 23.3 TB/s; step 3 (134M v_exp_f32 + WMMA) is the compute phase.

#define HW_  4096
#define C_   256
#define B_   8

typedef __attribute__((ext_vector_type(2))) float v2f;
typedef __attribute__((ext_vector_type(8))) float v8f;

// ---------------------------------------------------------------------------
// Kernel 1: att[b,p] = sum_c x[b,c,p] * w_map[c] + b_map
// Coalesced across p (stride-4096 over c); w_map cached in LDS.
// ---------------------------------------------------------------------------
__global__ __launch_bounds__(256) void k_channel_reduce(
    const float* __restrict__ x, const float* __restrict__ w_map,
    const float* __restrict__ b_map, float* __restrict__ att)
{
  __shared__ float wm[C_];
  const int tid = threadIdx.x;
  wm[tid] = w_map[tid];            // 256 threads == 256 channels
  __syncthreads();

  const int b = blockIdx.x >> 4;               // 16 blocks per batch
  const int p = ((blockIdx.x & 15) << 8) + tid;
  const float* xp = x + ((size_t)b * C_) * HW_ + p;

  float a0 = 0.f, a1 = 0.f, a2 = 0.f, a3 = 0.f;
  #pragma unroll 4
  for (int c = 0; c < C_; c += 4) {
    a0 = fmaf(xp[(size_t)(c + 0) * HW_], wm[c + 0], a0);
    a1 = fmaf(xp[(size_t)(c + 1) * HW_], wm[c + 1], a1);
    a2 = fmaf(xp[(size_t)(c + 2) * HW_], wm[c + 2], a2);
    a3 = fmaf(xp[(size_t)(c + 3) * HW_], wm[c + 3], a3);
  }
  att[b * HW_ + p] = (a0 + a1) + (a2 + a3) + b_map[0];
}

// ---------------------------------------------------------------------------
// Kernel 2: per-batch min/max of att (feeds the softmax row-max).
// ---------------------------------------------------------------------------
__global__ __launch_bounds__(256) void k_minmax(
    const float* __restrict__ att, float* __restrict__ amin,
    float* __restrict__ amax)
{
  __shared__ float slo[256], shi[256];
  const int b = blockIdx.x, tid = threadIdx.x;
  const float* a = att + b * HW_;
  float lo = 3.0e38f, hi = -3.0e38f;
  for (int i = tid; i < HW_; i += 256) {
    const float v = a[i];
    lo = fminf(lo, v);
    hi = fmaxf(hi, v);
  }
  slo[tid] = lo; shi[tid] = hi;
  __syncthreads();
  for (int s = 128; s > 0; s >>= 1) {
    if (tid < s) {
      slo[tid] = fminf(slo[tid], slo[tid + s]);
      shi[tid] = fmaxf(shi[tid], shi[tid + s]);
    }
    __syncthreads();
  }
  if (tid == 0) { amin[b] = slo[0]; amax[b] = shi[0]; }
}

// ---------------------------------------------------------------------------
// Kernel 3: rank-1 softmax rows via WMMA.
//   gate[b,i] = softsign( num_i / den_i )
//   num_i = sum_j exp(a_i*att_j + c_i) * att_j,  den_i = sum_j exp(a_i*att_j + c_i)
// with a_i = q_i*w_k and c_i = q_i*b_k - m_i (m_i = row max of q_i*k_j).
//
// Workgroup = 256 threads = 8 waves, handles 2 row-tiles of 16 rows each
// (4 waves per tile split over the j dimension, partials merged in LDS).
// Per WMMA: A (16x4 f32) = exp tile; B (4x16 f32) col0 = att_j, col1 = 1.
// ---------------------------------------------------------------------------
__global__ __launch_bounds__(256) void k_rank1_softmax_wmma(
    const float* __restrict__ att,
    const float* __restrict__ amin, const float* __restrict__ amax,
    const float* __restrict__ w_q_p, const float* __restrict__ b_q_p,
    const float* __restrict__ w_k_p, const float* __restrict__ b_k_p,
    float* __restrict__ gate)
{
  __shared__ float att_s[HW_];           // 16 KB: whole batch row of att
  __shared__ float red[2][4][16][2];     // [tile][jseg][row][num/den]

  const int tid  = threadIdx.x;
  const int b    = blockIdx.x >> 7;      // 128 workgroups per batch
  const int wgid = blockIdx.x & 127;

  // Stage att[b,:] into LDS (float4, fully coalesced).
  {
    const float4* g4 = (const float4*)(att + b * HW_);
    float4* s4 = (float4*)att_s;
    for (int i = tid; i < HW_ / 4; i += 256) s4[i] = g4[i];
  }
  __syncthreads();

  const int wave = tid >> 5;
  const int lane = tid & 31;
  const int t2   = wave >> 2;            // tile within workgroup (0/1)
  const int wseg = wave & 3;             // j-segment handled by this wave
  const int tile = wgid * 2 + t2;        // global 16-row tile id in [0,256)
  const int M    = lane & 15;            // row within tile (A row; B and D column)
  const int half = lane >> 4;            // K-half select
  const int koff = half * 2;

  const float w_q = w_q_p[0], b_q = b_q_p[0];
  const float w_k = w_k_p[0], b_k = b_k_p[0];

  const float att_i = att_s[tile * 16 + M];
  const float q_i   = fmaf(w_q, att_i, b_q);
  const float khi   = fmaf(w_k, amax[b], b_k);
  const float klo   = fmaf(w_k, amin[b], b_k);
  const float m_i   = fmaxf(q_i * khi, q_i * klo);   // max_j q_i*k_j
  const float a_i   = q_i * w_k;
  const float c_i   = fmaf(q_i, b_k, -m_i);

  const bool isc0 = (M == 0), isc1 = (M == 1);

  v8f acc = {};
  #pragma unroll 4
  for (int it = 0; it < 256; ++it) {
    const int jb = wseg * 4 + it * 16;             // 4 j-columns per WMMA
    const float av0 = att_s[jb + koff];            // broadcast LDS reads
    const float av1 = att_s[jb + koff + 1];

    v2f A, Bm;
    A.x = __expf(fmaf(a_i, av0, c_i));             // E[M][K]   (<= 1.0)
    A.y = __expf(fmaf(a_i, av1, c_i));
    Bm.x = isc0 ? av0 : (isc1 ? 1.0f : 0.0f);      // B[K][0]=att, B[K][1]=1
    Bm.y = isc0 ? av1 : (isc1 ? 1.0f : 0.0f);

    // D = A*B + C : col0 accumulates sum(E*att), col1 accumulates sum(E)
    acc = __builtin_amdgcn_wmma_f32_16x16x4_f32(
        /*neg_a=*/false, A, /*neg_b=*/false, Bm,
        /*c_mod=*/(short)0, acc, /*reuse_a=*/false, /*reuse_b=*/false);
  }

  // Extract columns 0/1 of the 16x16 accumulator: VGPR v holds row v+8*half,
  // lane&15 is the column. Lanes 0/16 own numerators, lanes 1/17 denominators.
  float cc[8];
  #pragma unroll
  for (int v = 0; v < 8; ++v) cc[v] = acc[v];
  #pragma unroll
  for (int v = 0; v < 8; ++v) {
    const int row = v + 8 * half;
    if (M == 0)      red[t2][wseg][row][0] = cc[v];
    else if (M == 1) red[t2][wseg][row][1] = cc[v];
  }
  __syncthreads();

  // Merge the 4 j-segments, finish softmax + softsign gate.
  if (tid < 32) {
    const int tt = tid >> 4, r = tid & 15;
    const float num = red[tt][0][r][0] + red[tt][1][r][0] +
                      red[tt][2][r][0] + red[tt][3][r][0];
    const float den = red[tt][0][r][1] + red[tt][1][r][1] +
                      red[tt][2][r][1] + red[tt][3][r][1];
    const float o = num / den;                     // den >= 1 (max-shifted)
    gate[b * HW_ + (wgid * 2 + tt) * 16 + r] = o / (1.0f + fabsf(o));
  }
}

// ---------------------------------------------------------------------------
// Kernel 4: out[b,c,p] = gate[b,p] * x[b,c,p]  (float4, gate hits L2/L0)
// ---------------------------------------------------------------------------
__global__ __launch_bounds__(256) void k_gate_mul(
    const float* __restrict__ x, const float* __restrict__ gate,
    float* __restrict__ out)
{
  const size_t i4  = (size_t)blockIdx.x * 256 + threadIdx.x; // float4 index
  const size_t lin = i4 * 4;
  const int b = (int)(lin >> 20);       // 256*4096 = 2^20 elements per batch
  const int p = (int)(lin & (HW_ - 1));

  const float4 xv = ((const float4*)x)[i4];
  const float4 gv = *(const float4*)(gate + b * HW_ + p);
  float4 o;
  o.x = xv.x * gv.x; o.y = xv.y * gv.y;
  o.z = xv.z * gv.z; o.w = xv.w * gv.w;
  ((float4*)out)[i4] = o;
}

// ---------------------------------------------------------------------------
// Launch
// ---------------------------------------------------------------------------
extern "C" void kernel_launch(void* const* d_in, const int* in_sizes, int n_in,
                              void* d_out, int out_size, void* d_ws, size_t ws_size,
                              hipStream_t stream) {
  const float* x     = (const float*)d_in[0];
  const float* w_map = (const float*)d_in[1];
  const float* b_map = (const float*)d_in[2];
  const float* w_q   = (const float*)d_in[3];
  const float* b_q   = (const float*)d_in[4];
  const float* w_k   = (const float*)d_in[5];
  const float* b_k   = (const float*)d_in[6];

  float* ws   = (float*)d_ws;
  float* att  = ws;                 // 32768 floats
  float* amin = ws + 32768;         // 8
  float* amax = ws + 32776;         // 8
  float* gate = ws + 32784;         // 32768 (byte offset 131136, 16B aligned)
  float* out  = (float*)d_out;

  k_channel_reduce<<<B_ * (HW_ / 256), 256, 0, stream>>>(x, w_map, b_map, att);
  k_minmax<<<B_, 256, 0, stream>>>(att, amin, amax);
  k_rank1_softmax_wmma<<<B_ * 128, 256, 0, stream>>>(att, amin, amax,
                                                     w_q, b_q, w_k, b_k, gate);
  k_gate_mul<<<(B_ * C_ * HW_) / 4 / 256, 256, 0, stream>>>(x, gate, out);
}